// DomainBatchNorm_9706626089249
// MI455X (gfx1250) — compile-verified
//
#include <hip/hip_runtime.h>

// DomainBatchNorm, MI455X (gfx1250, wave32).
// out = x * (mask @ scale) + (mask @ shift), scale/shift folded per-domain.
// Memory-bound: ~258 MiB total traffic -> ~11 us @ 23.3 TB/s. The (Bx8)x(8xF)
// mask contraction is done per 16x16 tile with V_WMMA_F32_16X16X4_F32 (K=8 via
// two chained WMMAs), eff tiles live only in registers (WMMA C layout), fused
// with the elementwise apply so x/out stream exactly once.

#define F_DIM   1024
#define B_ROWS  32768
#define D_DOM   8
#define EPSBN   1e-5f

typedef __attribute__((ext_vector_type(2))) float v2f;
typedef __attribute__((ext_vector_type(8))) float v8f;

// ---------------------------------------------------------------------------
// Kernel 1: fold per-domain params into scale/shift tables (D*F = 8192 elems).
// scaleT[d][f] = gamma * rsqrt(var + eps); shiftT[d][f] = beta - mean * scale.
// ---------------------------------------------------------------------------
__global__ __launch_bounds__(256) void dbn_prep(const float* __restrict__ gammas,
                                                const float* __restrict__ betas,
                                                const float* __restrict__ pop_means,
                                                const float* __restrict__ pop_vars,
                                                float* __restrict__ scaleT,
                                                float* __restrict__ shiftT) {
    const int i = blockIdx.x * blockDim.x + threadIdx.x;   // grid sized exactly to D*F
    const float s = gammas[i] * __frsqrt_rn(pop_vars[i] + EPSBN);
    scaleT[i] = s;
    shiftT[i] = betas[i] - pop_means[i] * s;
}

// ---------------------------------------------------------------------------
// Kernel 2: fused WMMA + apply. One wave handles 16 rows x 64 cols
// (4 col-tiles). Per tile: 4x v_wmma_f32_16x16x4_f32 build eff_scale/eff_shift
// 16x16 tiles in registers, then 8 b32 loads of x, 8 fmas, 8 b32 stores.
//
// WMMA f32 operand layouts (wave32):
//   A 16x4 : lane&15 = M;  VGPR0 -> K = 2*(lane>>4), VGPR1 -> K+1
//   B 4x16 : lane&15 = N;  VGPR0 -> K = 2*(lane>>4), VGPR1 -> K+1
//   C 16x16: VGPR r -> M = r + 8*(lane>>4), N = lane&15
// ---------------------------------------------------------------------------
__global__ __launch_bounds__(256) void dbn_main(const float* __restrict__ x,
                                                const float* __restrict__ mask,
                                                const float* __restrict__ scaleT,
                                                const float* __restrict__ shiftT,
                                                float* __restrict__ out) {
    const int lane = threadIdx.x & 31;
    const int wave = threadIdx.x >> 5;
    const int gw   = blockIdx.x * 8 + wave;          // global wave id

    const int colGroups = F_DIM / 64;                // 16 groups of 64 columns
    const int tileM = gw / colGroups;                // 0 .. B/16-1
    const int cgrp  = gw % colGroups;

    const int r0 = tileM * 16;                       // first row of tile
    const int n  = lane & 15;                        // N (col) / M (row for A)
    const int kh = lane >> 4;                        // K half selector

    // --- A operand: mask tile 16x8, loaded once per wave (two float2 pairs) ---
    const float* mrow = mask + (size_t)(r0 + n) * D_DOM;
    v2f a0, a1;
    {
        const float2 p0 = *reinterpret_cast<const float2*>(mrow + 2 * kh);      // K 0..3
        const float2 p1 = *reinterpret_cast<const float2*>(mrow + 4 + 2 * kh);  // K 4..7
        a0.x = p0.x; a0.y = p0.y;
        a1.x = p1.x; a1.y = p1.y;
    }

    const int krow0 = 2 * kh;        // K rows this lane supplies for B (first WMMA)
    const int krow1 = 4 + 2 * kh;    // second WMMA (K=4..7)

#pragma unroll
    for (int t = 0; t < 4; ++t) {
        const int col = cgrp * 64 + t * 16 + n;

        // --- B operands: scale/shift tile columns, K rows strided by F ---
        v2f bs0, bs1, bh0, bh1;
        bs0.x = scaleT[(size_t)(krow0    ) * F_DIM + col];
        bs0.y = scaleT[(size_t)(krow0 + 1) * F_DIM + col];
        bs1.x = scaleT[(size_t)(krow1    ) * F_DIM + col];
        bs1.y = scaleT[(size_t)(krow1 + 1) * F_DIM + col];
        bh0.x = shiftT[(size_t)(krow0    ) * F_DIM + col];
        bh0.y = shiftT[(size_t)(krow0 + 1) * F_DIM + col];
        bh1.x = shiftT[(size_t)(krow1    ) * F_DIM + col];
        bh1.y = shiftT[(size_t)(krow1 + 1) * F_DIM + col];

        // --- eff_scale / eff_shift 16x16 tiles via chained K=4 WMMAs ---
        v8f cs = {};
        cs = __builtin_amdgcn_wmma_f32_16x16x4_f32(false, a0, false, bs0,
                                                   (short)0, cs, false, false);
        cs = __builtin_amdgcn_wmma_f32_16x16x4_f32(false, a1, false, bs1,
                                                   (short)0, cs, false, false);
        v8f ch = {};
        ch = __builtin_amdgcn_wmma_f32_16x16x4_f32(false, a0, false, bh0,
                                                   (short)0, ch, false, false);
        ch = __builtin_amdgcn_wmma_f32_16x16x4_f32(false, a1, false, bh1,
                                                   (short)0, ch, false, false);

        // --- fused apply on the WMMA C layout: out = x*eff_scale + eff_shift ---
        const size_t base = (size_t)(r0 + 8 * kh) * F_DIM + col;
        const float* xcol = x + base;
        float*       ocol = out + base;
#pragma unroll
        for (int r = 0; r < 8; ++r) {
            const float xv = xcol[(size_t)r * F_DIM];
            ocol[(size_t)r * F_DIM] = fmaf(xv, cs[r], ch[r]);
        }
    }
}

extern "C" void kernel_launch(void* const* d_in, const int* in_sizes, int n_in,
                              void* d_out, int out_size, void* d_ws, size_t ws_size,
                              hipStream_t stream) {
    // setup_inputs() order: inputs, mask, gammas, betas, pop_means, pop_vars
    const float* inputs    = (const float*)d_in[0];
    const float* mask      = (const float*)d_in[1];
    const float* gammas    = (const float*)d_in[2];
    const float* betas     = (const float*)d_in[3];
    const float* pop_means = (const float*)d_in[4];
    const float* pop_vars  = (const float*)d_in[5];
    float* out = (float*)d_out;

    float* scaleT = (float*)d_ws;                    // [D, F]
    float* shiftT = scaleT + (size_t)D_DOM * F_DIM;  // [D, F]  (64 KiB total)

    dbn_prep<<<(D_DOM * F_DIM) / 256, 256, 0, stream>>>(gammas, betas, pop_means,
                                                        pop_vars, scaleT, shiftT);

    const int totalWaves = (B_ROWS / 16) * (F_DIM / 64);  // 32768 waves
    dbn_main<<<totalWaves / 8, 256, 0, stream>>>(inputs, mask, scaleT, shiftT, out);
}